// GATBase_9818295239345
// MI455X (gfx1250) — compile-verified
//
#include <hip/hip_runtime.h>
#include <hip/hip_bf16.h>

// ---------------------------------------------------------------------------
// GAT 2-layer forward for MI455X (gfx1250, wave32).
// fp32 everywhere; GEMMs via V_WMMA_F32_16X16X4_F32.
// Edge gather vectorized to global_load_b128 on the bandwidth-critical pass.
// ---------------------------------------------------------------------------

#define GAT_N        50000
#define GAT_E        800000
#define GAT_FIN      128
#define GAT_H1       8
#define GAT_F1       16
#define GAT_FS1      128      // H1*F1
#define GAT_C        32
#define NEG_SLOPE    0.2f

typedef __attribute__((ext_vector_type(2))) float v2f;
typedef __attribute__((ext_vector_type(8))) float v8f;

__device__ __forceinline__ float lrelu(float x) {
  return x > 0.f ? x : NEG_SLOPE * x;
}

// ---------------------------------------------------------------------------
// GEMM: out[M x Nout] = A[M x K] @ W[K x Nout]   (row major, fp32)
// One wave computes a 16x32 output tile with two v8f accumulators.
// Requires M % 16 == 0, K % 4 == 0, Nout % 32 == 0. (50000 = 3125*16 ✓)
// ---------------------------------------------------------------------------
__global__ void gat_gemm_wmma_f32(const float* __restrict__ A,
                                  const float* __restrict__ W,
                                  float* __restrict__ out,
                                  int M, int K, int Nout) {
  const int lane = threadIdx.x & 31;
  const int wid  = threadIdx.x >> 5;
  const int wpb  = blockDim.x >> 5;
  const int half = lane >> 4;       // 0: K pair {k,k+1}, 1: {k+2,k+3}
  const int l16  = lane & 15;
  const int colGroups = Nout >> 5;  // 16x32 tiles horizontally
  const int ntiles = (M >> 4) * colGroups;

  for (int tile = blockIdx.x * wpb + wid; tile < ntiles;
       tile += (int)gridDim.x * wpb) {          // wave-uniform loop
    const int r0 = (tile / colGroups) << 4;
    const int n0 = (tile % colGroups) << 5;

    v8f acc0 = {}; v8f acc1 = {};
    const float* arow = A + (size_t)(r0 + l16) * K + (half << 1);
    const float* wcol = W + (size_t)(half << 1) * Nout + n0 + l16;

    for (int k = 0; k < K; k += 4) {
      // A fragment 16x4 f32: lane<16 holds A[r0+l16][k..k+1], lane>=16 [k+2..k+3]
      v2f a;
      { const float2 t = *(const float2*)(arow + k); a.x = t.x; a.y = t.y; }
      // B fragments 4x16 f32 (two adjacent 16-col tiles)
      const float* wp = wcol + (size_t)k * Nout;
      v2f b0, b1;
      b0.x = wp[0];   b0.y = wp[Nout];
      b1.x = wp[16];  b1.y = wp[Nout + 16];
      acc0 = __builtin_amdgcn_wmma_f32_16x16x4_f32(false, a, false, b0,
                                                   (short)0, acc0, false, false);
      acc1 = __builtin_amdgcn_wmma_f32_16x16x4_f32(false, a, false, b1,
                                                   (short)0, acc1, false, false);
    }
    // C/D layout: acc[i] = C[M = i + 8*half][N = l16]
    float* o = out + (size_t)(r0 + (half << 3)) * Nout + n0 + l16;
#pragma unroll
    for (int i = 0; i < 8; ++i) {
      o[(size_t)i * Nout]      = acc0[i];
      o[(size_t)i * Nout + 16] = acc1[i];
    }
  }
}

// ---------------------------------------------------------------------------
// Attention logits: al[n,h] = sum_f H[n, h*F+f] * a[h*F+f]   (src and dst)
// One wave per node. FS in {128, 32}; lanesPerHead = 32*F/FS.
// ---------------------------------------------------------------------------
__global__ void gat_att_logits(const float* __restrict__ H,
                               const float* __restrict__ aS,
                               const float* __restrict__ aD,
                               float* __restrict__ alS,
                               float* __restrict__ alD,
                               int Nn, int FS, int Hh, int lanesPerHead) {
  const int node = blockIdx.x * (blockDim.x >> 5) + (threadIdx.x >> 5);
  if (node >= Nn) return;                       // wave-uniform
  const int lane = threadIdx.x & 31;
  const int per = FS >> 5;
  const float* hr = H + (size_t)node * FS + lane * per;
  float ps = 0.f, pd = 0.f;
  for (int j = 0; j < per; ++j) {
    const float hv = hr[j];
    ps += hv * aS[lane * per + j];
    pd += hv * aD[lane * per + j];
  }
  for (int off = 1; off < lanesPerHead; off <<= 1) {
    ps += __shfl_xor(ps, off, 32);
    pd += __shfl_xor(pd, off, 32);
  }
  if ((lane & (lanesPerHead - 1)) == 0) {
    const int h = lane / lanesPerHead;
    alS[(size_t)node * Hh + h] = ps;
    alD[(size_t)node * Hh + h] = pd;
  }
}

// m[n,h] = e_self = lrelu(alS[n,h] + alD[n,h])   (self-loop seeds the max)
__global__ void gat_init_max(const float* __restrict__ alS,
                             const float* __restrict__ alD,
                             float* __restrict__ m, int NH) {
  const int i = blockIdx.x * blockDim.x + threadIdx.x;
  if (i < NH) m[i] = lrelu(alS[i] + alD[i]);
}

__global__ void gat_edge_max(const int* __restrict__ src,
                             const int* __restrict__ dst,
                             const float* __restrict__ alS,
                             const float* __restrict__ alD,
                             float* __restrict__ m, int Ecnt, int Hh) {
  const int e = blockIdx.x * blockDim.x + threadIdx.x;
  if (e >= Ecnt) return;
  const int s = src[e], d = dst[e];
  for (int h = 0; h < Hh; ++h) {
    const float v = lrelu(alS[(size_t)s * Hh + h] + alD[(size_t)d * Hh + h]);
    atomicMax(&m[(size_t)d * Hh + h], v);       // global_atomic_max_num_f32
  }
}

// denom[n,h] = exp(e_self - m)   (self-loop term)
__global__ void gat_init_denom(const float* __restrict__ alS,
                               const float* __restrict__ alD,
                               const float* __restrict__ m,
                               float* __restrict__ dn, int NH) {
  const int i = blockIdx.x * blockDim.x + threadIdx.x;
  if (i < NH) dn[i] = __expf(lrelu(alS[i] + alD[i]) - m[i]);
}

__global__ void gat_edge_sum(const int* __restrict__ src,
                             const int* __restrict__ dst,
                             const float* __restrict__ alS,
                             const float* __restrict__ alD,
                             const float* __restrict__ m,
                             float* __restrict__ dn, int Ecnt, int Hh) {
  const int e = blockIdx.x * blockDim.x + threadIdx.x;
  if (e >= Ecnt) return;
  const int s = src[e], d = dst[e];
  for (int h = 0; h < Hh; ++h) {
    const float v = lrelu(alS[(size_t)s * Hh + h] + alD[(size_t)d * Hh + h]);
    atomicAdd(&dn[(size_t)d * Hh + h], __expf(v - m[(size_t)d * Hh + h]));
  }
}

// out[n, 4f..4f+3] = alpha_self * H[n, 4f..4f+3]  (full overwrite, float4)
// F (features/head) is a multiple of 4 for both layers, so one float4 stays
// inside a single head.
__global__ void gat_init_out4(const float* __restrict__ H,
                              const float* __restrict__ alS,
                              const float* __restrict__ alD,
                              const float* __restrict__ m,
                              const float* __restrict__ dn,
                              float* __restrict__ out,
                              int Nn, int FS, int F, int Hh) {
  const int q = FS >> 2;                         // float4s per row
  const int i = blockIdx.x * blockDim.x + threadIdx.x;
  if (i >= Nn * q) return;
  const int n  = i / q;
  const int f4 = i - n * q;
  const int h  = (f4 << 2) / F;
  const float v = lrelu(alS[(size_t)n * Hh + h] + alD[(size_t)n * Hh + h]);
  const float alpha = __expf(v - m[(size_t)n * Hh + h]) / dn[(size_t)n * Hh + h];
  const float4 hv = ((const float4*)H)[i];
  float4 o;
  o.x = alpha * hv.x; o.y = alpha * hv.y; o.z = alpha * hv.z; o.w = alpha * hv.w;
  ((float4*)out)[i] = o;
}

// One wave per edge: out[d,:] += alpha * H[s,:]
// PER = floats per lane (4 for layer 1, 1 for layer 2).
template <int PER>
__global__ void gat_edge_agg(const int* __restrict__ src,
                             const int* __restrict__ dst,
                             const float* __restrict__ H,
                             const float* __restrict__ alS,
                             const float* __restrict__ alD,
                             const float* __restrict__ m,
                             const float* __restrict__ dn,
                             float* __restrict__ out,
                             int Ecnt, int F, int Hh) {
  const int e = blockIdx.x * (blockDim.x >> 5) + (threadIdx.x >> 5);
  if (e >= Ecnt) return;                        // wave-uniform
  const int lane = threadIdx.x & 31;
  const int s = src[e], d = dst[e];
  const int FS = PER * 32;
  const int f0 = lane * PER;
  const int h  = f0 / F;
  const float v = lrelu(alS[(size_t)s * Hh + h] + alD[(size_t)d * Hh + h]);
  const float alpha = __expf(v - m[(size_t)d * Hh + h]) / dn[(size_t)d * Hh + h];
  const float* hp = H + (size_t)s * FS + f0;
  float* op = out + (size_t)d * FS + f0;
  if constexpr (PER == 4) {
    const float4 hv = *(const float4*)hp;       // global_load_b128 gather
    atomicAdd(&op[0], alpha * hv.x);
    atomicAdd(&op[1], alpha * hv.y);
    atomicAdd(&op[2], alpha * hv.z);
    atomicAdd(&op[3], alpha * hv.w);
  } else {
#pragma unroll
    for (int j = 0; j < PER; ++j)
      atomicAdd(&op[j], alpha * hp[j]);
  }
}

// out = elu(out + b), vectorized float4 (FS % 4 == 0)
__global__ void gat_elu_bias4(float* __restrict__ out,
                              const float* __restrict__ b, int total4, int FS) {
  const int i = blockIdx.x * blockDim.x + threadIdx.x;
  if (i >= total4) return;
  const int q  = FS >> 2;
  const int f4 = i % q;
  const float4 bv = ((const float4*)b)[f4];
  float4 v = ((const float4*)out)[i];
  v.x += bv.x; v.y += bv.y; v.z += bv.z; v.w += bv.w;
  v.x = v.x > 0.f ? v.x : (__expf(v.x) - 1.f);
  v.y = v.y > 0.f ? v.y : (__expf(v.y) - 1.f);
  v.z = v.z > 0.f ? v.z : (__expf(v.z) - 1.f);
  v.w = v.w > 0.f ? v.w : (__expf(v.w) - 1.f);
  ((float4*)out)[i] = v;
}

// Per-node log_softmax over 32 classes: one wave, one class per lane.
__global__ void gat_bias_logsoftmax32(float* __restrict__ out,
                                      const float* __restrict__ b, int Nn) {
  const int node = blockIdx.x * (blockDim.x >> 5) + (threadIdx.x >> 5);
  if (node >= Nn) return;                       // wave-uniform
  const int lane = threadIdx.x & 31;
  const float x = out[(size_t)node * 32 + lane] + b[lane];
  float mx = x;
  for (int off = 1; off < 32; off <<= 1) mx = fmaxf(mx, __shfl_xor(mx, off, 32));
  const float ex = __expf(x - mx);
  float sm = ex;
  for (int off = 1; off < 32; off <<= 1) sm += __shfl_xor(sm, off, 32);
  out[(size_t)node * 32 + lane] = x - mx - __logf(sm);
}

// ---------------------------------------------------------------------------
static inline int cdiv(int a, int b) { return (a + b - 1) / b; }

extern "C" void kernel_launch(void* const* d_in, const int* in_sizes, int n_in,
                              void* d_out, int out_size, void* d_ws, size_t ws_size,
                              hipStream_t stream) {
  (void)in_sizes; (void)n_in; (void)out_size; (void)ws_size;
  const float* x     = (const float*)d_in[0];
  const int*   src   = (const int*)  d_in[1];
  const int*   dst   = (const int*)  d_in[2];
  const float* W1    = (const float*)d_in[3];
  const float* aS1   = (const float*)d_in[4];
  const float* aD1   = (const float*)d_in[5];
  const float* b1    = (const float*)d_in[6];
  const float* W2    = (const float*)d_in[7];
  const float* aS2   = (const float*)d_in[8];
  const float* aD2   = (const float*)d_in[9];
  const float* b2    = (const float*)d_in[10];
  float* outp = (float*)d_out;

  // workspace layout (floats)
  float* ws   = (float*)d_ws;
  float* H1m  = ws;                                  // N*128
  float* OUT1 = H1m  + (size_t)GAT_N * GAT_FS1;      // N*128
  float* H2m  = OUT1 + (size_t)GAT_N * GAT_FS1;      // N*32
  float* als1 = H2m  + (size_t)GAT_N * GAT_C;        // N*8
  float* ald1 = als1 + (size_t)GAT_N * GAT_H1;       // N*8
  float* m1   = ald1 + (size_t)GAT_N * GAT_H1;       // N*8
  float* dn1  = m1   + (size_t)GAT_N * GAT_H1;       // N*8
  float* als2 = dn1  + (size_t)GAT_N * GAT_H1;       // N
  float* ald2 = als2 + (size_t)GAT_N;                // N
  float* m2   = ald2 + (size_t)GAT_N;                // N
  float* dn2  = m2   + (size_t)GAT_N;                // N

  const int BLK = 256;               // 8 waves per block
  const int WPB = BLK / 32;

  // ---------------- Layer 1 ----------------
  {
    const int tiles = (GAT_N / 16) * (GAT_FS1 / 32);
    gat_gemm_wmma_f32<<<cdiv(tiles, WPB), BLK, 0, stream>>>(
        x, W1, H1m, GAT_N, GAT_FIN, GAT_FS1);

    gat_att_logits<<<cdiv(GAT_N, WPB), BLK, 0, stream>>>(
        H1m, aS1, aD1, als1, ald1, GAT_N, GAT_FS1, GAT_H1, /*lanesPerHead=*/4);

    const int NH = GAT_N * GAT_H1;
    gat_init_max<<<cdiv(NH, BLK), BLK, 0, stream>>>(als1, ald1, m1, NH);
    gat_edge_max<<<cdiv(GAT_E, BLK), BLK, 0, stream>>>(
        src, dst, als1, ald1, m1, GAT_E, GAT_H1);
    gat_init_denom<<<cdiv(NH, BLK), BLK, 0, stream>>>(als1, ald1, m1, dn1, NH);
    gat_edge_sum<<<cdiv(GAT_E, BLK), BLK, 0, stream>>>(
        src, dst, als1, ald1, m1, dn1, GAT_E, GAT_H1);

    const int tot4 = GAT_N * (GAT_FS1 / 4);
    gat_init_out4<<<cdiv(tot4, BLK), BLK, 0, stream>>>(
        H1m, als1, ald1, m1, dn1, OUT1, GAT_N, GAT_FS1, GAT_F1, GAT_H1);
    gat_edge_agg<4><<<cdiv(GAT_E, WPB), BLK, 0, stream>>>(
        src, dst, H1m, als1, ald1, m1, dn1, OUT1, GAT_E, GAT_F1, GAT_H1);

    gat_elu_bias4<<<cdiv(tot4, BLK), BLK, 0, stream>>>(OUT1, b1, tot4, GAT_FS1);
  }

  // ---------------- Layer 2 ----------------
  {
    const int tiles = (GAT_N / 16) * (GAT_C / 32);
    gat_gemm_wmma_f32<<<cdiv(tiles, WPB), BLK, 0, stream>>>(
        OUT1, W2, H2m, GAT_N, GAT_FS1, GAT_C);

    gat_att_logits<<<cdiv(GAT_N, WPB), BLK, 0, stream>>>(
        H2m, aS2, aD2, als2, ald2, GAT_N, GAT_C, 1, /*lanesPerHead=*/32);

    gat_init_max<<<cdiv(GAT_N, BLK), BLK, 0, stream>>>(als2, ald2, m2, GAT_N);
    gat_edge_max<<<cdiv(GAT_E, BLK), BLK, 0, stream>>>(
        src, dst, als2, ald2, m2, GAT_E, 1);
    gat_init_denom<<<cdiv(GAT_N, BLK), BLK, 0, stream>>>(als2, ald2, m2, dn2, GAT_N);
    gat_edge_sum<<<cdiv(GAT_E, BLK), BLK, 0, stream>>>(
        src, dst, als2, ald2, m2, dn2, GAT_E, 1);

    const int tot4 = GAT_N * (GAT_C / 4);
    gat_init_out4<<<cdiv(tot4, BLK), BLK, 0, stream>>>(
        H2m, als2, ald2, m2, dn2, outp, GAT_N, GAT_C, GAT_C, 1);
    gat_edge_agg<1><<<cdiv(GAT_E, WPB), BLK, 0, stream>>>(
        src, dst, H2m, als2, ald2, m2, dn2, outp, GAT_E, GAT_C, 1);

    gat_bias_logsoftmax32<<<cdiv(GAT_N, WPB), BLK, 0, stream>>>(outp, b2, GAT_N);
  }
}